// Rasterer_63960652972629
// MI455X (gfx1250) — compile-verified
//
#include <hip/hip_runtime.h>
#include <hip/hip_bf16.h>

#define RES_X 128
#define RES_Y 128
#define NPTS  1024
#define DIAM  0.04f
#define SLOPE 400.0f
#define BETA  50.0f
#define NPIX  (RES_X * RES_Y)

typedef __attribute__((ext_vector_type(2))) float v2f;
typedef __attribute__((ext_vector_type(8))) float v8f;

// One wave = 16 pixels. A = point tile (M=16 points, K=4 padded),
// B = pixel rays (K=4 x N=16 pixels). D[m][n] = dot(point_m, ray_n).
// D layout: lane = n + 16*(m>=8), vgpr v = m%8  -> per lane: 8 points, 1 pixel.
__global__ __launch_bounds__(256) void Rasterer_63960652972629_kernel(
    const float* __restrict__ coords,   // [1024,3]
    const float* __restrict__ normals,  // [1024,3]
    const float* __restrict__ colors,   // [1024,3]
    const float* __restrict__ cam,      // [7] quat(wxyz) + trans
    const float* __restrict__ Kmat,     // [3,3]
    float* __restrict__ out)            // [3,128,128]
{
  __shared__ float s_p3x[NPTS], s_p3y[NPTS], s_p3z[NPTS];
  __shared__ float s_n3x[NPTS], s_n3y[NPTS], s_n3z[NPTS];
  __shared__ float s_num[NPTS], s_p2[NPTS];
  __shared__ float s_cr[NPTS], s_cg[NPTS], s_cb[NPTS];

  // ---- quaternion -> DCM ----
  float qw = cam[0], qx = cam[1], qy = cam[2], qz = cam[3];
  float qn = rsqrtf(qw*qw + qx*qx + qy*qy + qz*qz);
  qw *= qn; qx *= qn; qy *= qn; qz *= qn;
  float R00 = 1.f - 2.f*(qy*qy + qz*qz), R01 = 2.f*(qx*qy - qw*qz), R02 = 2.f*(qx*qz + qw*qy);
  float R10 = 2.f*(qx*qy + qw*qz), R11 = 1.f - 2.f*(qx*qx + qz*qz), R12 = 2.f*(qy*qz - qw*qx);
  float R20 = 2.f*(qx*qz - qw*qy), R21 = 2.f*(qy*qz + qw*qx), R22 = 1.f - 2.f*(qx*qx + qy*qy);
  float tx = cam[4], ty = cam[5], tz = cam[6];

  // ---- transform all points into LDS (SoA), 4 per thread ----
  int tid = threadIdx.x;
  for (int i = tid; i < NPTS; i += 256) {
    float cx = coords[3*i], cy = coords[3*i+1], cz = coords[3*i+2];
    float nx = normals[3*i], ny = normals[3*i+1], nz = normals[3*i+2];
    float px = R00*cx + R01*cy + R02*cz + tx;
    float py = R10*cx + R11*cy + R12*cz + ty;
    float pz = R20*cx + R21*cy + R22*cz + tz;
    float mx = R00*nx + R01*ny + R02*nz;
    float my = R10*nx + R11*ny + R12*nz;
    float mz = R20*nx + R21*ny + R22*nz;
    s_p3x[i] = px; s_p3y[i] = py; s_p3z[i] = pz;
    s_n3x[i] = mx; s_n3y[i] = my; s_n3z[i] = mz;
    s_num[i] = px*mx + py*my + pz*mz;
    s_p2[i]  = px*px + py*py + pz*pz;
    s_cr[i] = colors[3*i]; s_cg[i] = colors[3*i+1]; s_cb[i] = colors[3*i+2];
  }
  __syncthreads();

  // ---- Kinv via adjugate ----
  float a = Kmat[0], b = Kmat[1], c = Kmat[2];
  float d = Kmat[3], e = Kmat[4], f = Kmat[5];
  float g = Kmat[6], h = Kmat[7], i9 = Kmat[8];
  float det = a*(e*i9 - f*h) - b*(d*i9 - f*g) + c*(d*h - e*g);
  float inv = 1.0f / det;
  float Ki00 = (e*i9 - f*h)*inv, Ki01 = (c*h - b*i9)*inv, Ki02 = (b*f - c*e)*inv;
  float Ki10 = (f*g - d*i9)*inv, Ki11 = (a*i9 - c*g)*inv, Ki12 = (c*d - a*f)*inv;
  float Ki20 = (d*h - e*g)*inv, Ki21 = (b*g - a*h)*inv, Ki22 = (a*e - b*d)*inv;

  int lane = tid & 31;
  int lo   = lane & 15;
  bool hi  = lane >= 16;
  int wave = tid >> 5;
  int tile = blockIdx.x * 8 + wave;   // 16-pixel tile, 0..1023
  int p    = tile * 16 + lo;          // pixel id = y*128 + x
  float fx = (float)(p & (RES_X - 1));
  float fy = (float)(p >> 7);
  float rx = Ki00*fx + Ki01*fy + Ki02;
  float ry = Ki10*fx + Ki11*fy + Ki12;
  float rz = Ki20*fx + Ki21*fy + Ki22;
  float r2 = rx*rx + ry*ry + rz*rz;

  // B operand: lanes 0-15 hold (K0,K1)=(x,y), lanes 16-31 hold (K2,K3)=(z,0)
  v2f bmat;
  bmat[0] = hi ? rz : rx;
  bmat[1] = hi ? 0.0f : ry;

  const v8f zeroC = {};

  // ================= pass 1: m = max over points of y = -BETA*tt ==========
  // (valid upper scale for logsumexp since log(w+1e-8) <= ~0)
  float mmax = -3.0e38f;
  for (int j = 0; j < NPTS / 16; ++j) {
    int idx = j * 16 + lo;
    v2f an;
    an[0] = hi ? s_n3z[idx] : s_n3x[idx];
    an[1] = hi ? 0.0f       : s_n3y[idx];
    v8f den8 = __builtin_amdgcn_wmma_f32_16x16x4_f32(false, an, false, bmat,
                                                     (short)0, zeroC, false, false);
    int mbase = j * 16 + (hi ? 8 : 0);
#pragma unroll
    for (int v = 0; v < 8; ++v) {
      float den = den8[v];
      den = (fabsf(den) > 1e-6f) ? den : 1e-6f;
      float tt = s_num[mbase + v] * __builtin_amdgcn_rcpf(den);
      mmax = fmaxf(mmax, -BETA * tt);
    }
  }
  mmax = fmaxf(mmax, __shfl_xor(mmax, 16, 32));

  // ================= pass 2: S = sum (w+1e-8) * exp(y - m) ================
  float ssum = 0.0f;
  for (int j = 0; j < NPTS / 16; ++j) {
    int idx = j * 16 + lo;
    v2f an, ap;
    an[0] = hi ? s_n3z[idx] : s_n3x[idx];
    an[1] = hi ? 0.0f       : s_n3y[idx];
    ap[0] = hi ? s_p3z[idx] : s_p3x[idx];
    ap[1] = hi ? 0.0f       : s_p3y[idx];
    v8f den8 = __builtin_amdgcn_wmma_f32_16x16x4_f32(false, an, false, bmat,
                                                     (short)0, zeroC, false, false);
    v8f rp8  = __builtin_amdgcn_wmma_f32_16x16x4_f32(false, ap, false, bmat,
                                                     (short)0, zeroC, false, false);
    int mbase = j * 16 + (hi ? 8 : 0);
#pragma unroll
    for (int v = 0; v < 8; ++v) {
      int m = mbase + v;
      float den = den8[v];
      den = (fabsf(den) > 1e-6f) ? den : 1e-6f;
      float tt = s_num[m] * __builtin_amdgcn_rcpf(den);
      float d2 = tt*tt*r2 - 2.0f*tt*rp8[v] + s_p2[m];
      float dist = __builtin_amdgcn_sqrtf(fmaxf(d2, 1e-12f));
      float ez = __expf(-SLOPE * (0.5f*DIAM - dist));
      float w = __builtin_amdgcn_rcpf(1.0f + ez);
      w = (tt > 0.001f) ? w : 0.0f;
      ssum += (w + 1e-8f) * __expf(-BETA*tt - mmax);
    }
  }
  ssum += __shfl_xor(ssum, 16, 32);
  float invS = __builtin_amdgcn_rcpf(fmaxf(ssum, 1e-37f));

  // ================= pass 3: vis = min(1, exp(y-m)/S), accumulate colors ==
  float cr = 0.0f, cg = 0.0f, cb = 0.0f;
  for (int j = 0; j < NPTS / 16; ++j) {
    int idx = j * 16 + lo;
    v2f an, ap;
    an[0] = hi ? s_n3z[idx] : s_n3x[idx];
    an[1] = hi ? 0.0f       : s_n3y[idx];
    ap[0] = hi ? s_p3z[idx] : s_p3x[idx];
    ap[1] = hi ? 0.0f       : s_p3y[idx];
    v8f den8 = __builtin_amdgcn_wmma_f32_16x16x4_f32(false, an, false, bmat,
                                                     (short)0, zeroC, false, false);
    v8f rp8  = __builtin_amdgcn_wmma_f32_16x16x4_f32(false, ap, false, bmat,
                                                     (short)0, zeroC, false, false);
    int mbase = j * 16 + (hi ? 8 : 0);
#pragma unroll
    for (int v = 0; v < 8; ++v) {
      int m = mbase + v;
      float den = den8[v];
      den = (fabsf(den) > 1e-6f) ? den : 1e-6f;
      float tt = s_num[m] * __builtin_amdgcn_rcpf(den);
      float d2 = tt*tt*r2 - 2.0f*tt*rp8[v] + s_p2[m];
      float dist = __builtin_amdgcn_sqrtf(fmaxf(d2, 1e-12f));
      float ez = __expf(-SLOPE * (0.5f*DIAM - dist));
      float w = __builtin_amdgcn_rcpf(1.0f + ez);
      w = (tt > 0.001f) ? w : 0.0f;
      float vis = fminf(1.0f, __expf(-BETA*tt - mmax) * invS);
      float prob = w * vis;
      cr += prob * s_cr[m];
      cg += prob * s_cg[m];
      cb += prob * s_cb[m];
    }
  }
  cr += __shfl_xor(cr, 16, 32);
  cg += __shfl_xor(cg, 16, 32);
  cb += __shfl_xor(cb, 16, 32);

  if (!hi) {
    out[0 * NPIX + p] = fminf(cr, 1.0f);
    out[1 * NPIX + p] = fminf(cg, 1.0f);
    out[2 * NPIX + p] = fminf(cb, 1.0f);
  }
}

extern "C" void kernel_launch(void* const* d_in, const int* in_sizes, int n_in,
                              void* d_out, int out_size, void* d_ws, size_t ws_size,
                              hipStream_t stream) {
  const float* coords  = (const float*)d_in[0];
  const float* normals = (const float*)d_in[1];
  const float* colors  = (const float*)d_in[2];
  const float* cam     = (const float*)d_in[3];
  const float* Kmat    = (const float*)d_in[4];
  float* out = (float*)d_out;
  // 16384 pixels / (8 waves * 16 px) = 128 blocks of 256 threads
  Rasterer_63960652972629_kernel<<<128, 256, 0, stream>>>(
      coords, normals, colors, cam, Kmat, out);
  (void)in_sizes; (void)n_in; (void)out_size; (void)d_ws; (void)ws_size;
}